// RelativisticChargedParticleLagrangianDynamicalSystem_29695403885186
// MI455X (gfx1250) — compile-verified
//
#include <hip/hip_runtime.h>
#include <math.h>

typedef __attribute__((ext_vector_type(2))) float v2f;
typedef __attribute__((ext_vector_type(4))) float v4f;
typedef __attribute__((ext_vector_type(8))) float v8f;

__device__ __forceinline__ v8f wmma_f32_16x16x4(v2f a, v2f b, v8f c) {
  // 8 args: (neg_a, A, neg_b, B, c_mod, C, reuse_a, reuse_b)
  return __builtin_amdgcn_wmma_f32_16x16x4_f32(false, a, false, b, (short)0, c,
                                               false, false);
}

__device__ __forceinline__ float fast_rcp(float x) {
  return __builtin_amdgcn_rcpf(x);  // v_rcp_f32, ~1 ulp
}

__device__ __forceinline__ float fast_tanh(float x) {
  // tanh(x) = 1 - 2/(exp(2x)+1); exact saturation at +/-inf, ~1e-7 abs error.
  float e = __expf(2.0f * x);
  return 1.0f - 2.0f * fast_rcp(e + 1.0f);
}

// One wave processes 32 particles per iteration as two 16-particle WMMA tiles:
//   D(16x16) = A(16x4 weights) x B(4x16 per-particle columns)
// A rows [W;V;W;V] make every lane of a tile receive the full (W*vec, V*vec)
// of "its" column; merging tile A (lanes 0-15) and tile B (lanes 16-31) gives
// each lane exactly one particle, so all elementwise math runs at full yield.
// B-operands for the x/u GEMMs are direct lane-remapped b64 re-reads of y
// (L0 hits on the lines the b128 loads fetched); only the computed p/g
// operands need cross-half shfl_xor staging.
__global__ __launch_bounds__(256, 1) void relativistic_lagrangian_kernel(
    const float* __restrict__ y, const float* __restrict__ W,
    const float* __restrict__ V, float* __restrict__ out, int nTiles) {
  const int lane = threadIdx.x & 31;
  const int row  = lane & 15;          // A-matrix row / column-in-tile
  const bool hi  = (lane & 16) != 0;   // K-half: false->K=0,1  true->K=2,3
  const int k0   = hi ? 2 : 0;
  const int rr   = row & 3;

  // A operand for [Wx|Vx], [Wu|Vu]: rows [W0..3, V0..3, W0..3, V0..3]
  const float* wv = (row & 4) ? V : W;
  v2f a_wv;
  a_wv.x = wv[rr * 4 + k0];
  a_wv.y = wv[rr * 4 + k0 + 1];

  // A operands for W^T / V^T products; rows 4-7 and 12-15 zeroed so two
  // chained WMMAs leave F = W^T(c*g) + V^T g in C VGPRs 0..3 of every lane.
  v2f a_wt, a_vt;
  if (row & 4) {
    a_wt.x = 0.f; a_wt.y = 0.f;
    a_vt.x = 0.f; a_vt.y = 0.f;
  } else {
    a_wt.x = W[k0 * 4 + rr]; a_wt.y = W[(k0 + 1) * 4 + rr];
    a_vt.x = V[k0 * 4 + rr]; a_vt.y = V[(k0 + 1) * 4 + rr];
  }

  const int wavesPerBlock = blockDim.x >> 5;
  const int wave   = blockIdx.x * wavesPerBlock + (threadIdx.x >> 5);
  const int nWaves = gridDim.x * wavesPerBlock;
  const v8f zero8 = {0.f, 0.f, 0.f, 0.f, 0.f, 0.f, 0.f, 0.f};

  // Per-lane float offset of this lane's B-operand slot within a tile pair.
  const int bOffLane = row * 8 + k0;   // + tile*256 each iteration

  for (int tile = wave; tile < nTiles; tile += nWaves) {
    const int pid = tile * 32 + lane;       // this lane's particle
    const float* yp = y + (size_t)pid * 8;
    v4f xv = *(const v4f*)(yp);             // 1KB contiguous per wave
    v4f uv = *(const v4f*)(yp + 4);

    // Direct lane-remapped B-operands (pure cache hits, no shuffles):
    // tile A col n: lanes n / n+16 read (x0,x1)/(x2,x3) of particle base+n.
    const float* bp = y + (size_t)tile * 256 + bOffLane;
    v2f bxA = *(const v2f*)(bp);        // x half, tile A
    v2f buA = *(const v2f*)(bp + 4);    // u half, tile A
    v2f bxB = *(const v2f*)(bp + 128);  // x half, tile B (+16 particles)
    v2f buB = *(const v2f*)(bp + 132);  // u half, tile B

    v8f d1A = wmma_f32_16x16x4(a_wv, bxA, zero8);  // [Wx|Vx], tile A
    v8f d1B = wmma_f32_16x16x4(a_wv, bxB, zero8);  // [Wx|Vx], tile B
    v8f d2A = wmma_f32_16x16x4(a_wv, buA, zero8);  // [Wu|Vu], tile A
    v8f d2B = wmma_f32_16x16x4(a_wv, buB, zero8);  // [Wu|Vu], tile B

    // Each lane keeps its own particle's results.
    v8f d1 = hi ? d1B : d1A;
    v8f d2 = hi ? d2B : d2A;

    // A = tanh(Wx) + Vx ; c = 1 - tanh^2(Wx)
    float t0 = fast_tanh(d1[0]), t1 = fast_tanh(d1[1]);
    float t2 = fast_tanh(d1[2]), t3 = fast_tanh(d1[3]);
    float A0 = t0 + d1[4], A1 = t1 + d1[5], A2 = t2 + d1[6], A3 = t3 + d1[7];
    float c0 = 1.f - t0 * t0, c1 = 1.f - t1 * t1;
    float c2 = 1.f - t2 * t2, c3 = 1.f - t3 * t3;

    float S = uv.x * A0 + uv.y * A1 + uv.z * A2 + uv.w * A3;  // u . A

    // g = dL/dA = (0.5*eta + 0.1*S) * u   (eta = +1,-1,-1,-1)
    float kc0 = 0.5f + 0.1f * S;
    float kcn = -0.5f + 0.1f * S;
    float g0 = kc0 * uv.x, g1 = kcn * uv.y, g2 = kcn * uv.z, g3 = kcn * uv.w;

    // w = dA_dx^T u = c*(Wu) + Vu
    float w0 = c0 * d2[0] + d2[4];
    float w1 = c1 * d2[1] + d2[5];
    float w2 = c2 * d2[2] + d2[6];
    float w3 = c3 * d2[3] + d2[7];

    // Stage p = c*g and g across halves for the transposed GEMMs.
    float p0 = c0 * g0, p1 = c1 * g1, p2 = c2 * g2, p3 = c3 * g3;
    float sp0 = __shfl_xor(p0, 16, 32), sp1 = __shfl_xor(p1, 16, 32);
    float sp2 = __shfl_xor(p2, 16, 32), sp3 = __shfl_xor(p3, 16, 32);
    float sg0 = __shfl_xor(g0, 16, 32), sg1 = __shfl_xor(g1, 16, 32);
    float sg2 = __shfl_xor(g2, 16, 32), sg3 = __shfl_xor(g3, 16, 32);

    v2f bpA, bpB, bgA, bgB;
    bpA.x = hi ? sp2 : p0;   bpA.y = hi ? sp3 : p1;
    bpB.x = hi ? p2 : sp0;   bpB.y = hi ? p3 : sp1;
    bgA.x = hi ? sg2 : g0;   bgA.y = hi ? sg3 : g1;
    bgB.x = hi ? g2 : sg0;   bgB.y = hi ? g3 : sg1;

    // F = W^T (c*g) + V^T g, per tile, chained through C.
    v8f d3A = wmma_f32_16x16x4(a_wt, bpA, zero8);
    v8f d4A = wmma_f32_16x16x4(a_vt, bgA, d3A);
    v8f d3B = wmma_f32_16x16x4(a_wt, bpB, zero8);
    v8f d4B = wmma_f32_16x16x4(a_vt, bgB, d3B);

    float F0 = hi ? d4B[0] : d4A[0];
    float F1 = hi ? d4B[1] : d4A[1];
    float F2 = hi ? d4B[2] : d4A[2];
    float F3 = hi ? d4B[3] : d4A[3];

    float wu = w0 * uv.x + w1 * uv.y + w2 * uv.z + w3 * uv.w;

    // force = -0.2 x u^2 + F - (0.5 eta + 0.1 S) w - 0.1 (w.u) A
    float f0 = -0.2f * xv.x * uv.x * uv.x + F0 - kc0 * w0 - 0.1f * wu * A0;
    float f1 = -0.2f * xv.y * uv.y * uv.y + F1 - kcn * w1 - 0.1f * wu * A1;
    float f2 = -0.2f * xv.z * uv.z * uv.z + F2 - kcn * w2 - 0.1f * wu * A2;
    float f3 = -0.2f * xv.w * uv.w * uv.w + F3 - kcn * w3 - 0.1f * wu * A3;

    // J_uu = diag(eta + 0.2 x^2) + 0.1 A A^T ; Sherman-Morrison solve.
    float iD0 = fast_rcp(1.f + 0.2f * xv.x * xv.x);
    float iD1 = fast_rcp(-1.f + 0.2f * xv.y * xv.y);
    float iD2 = fast_rcp(-1.f + 0.2f * xv.z * xv.z);
    float iD3 = fast_rcp(-1.f + 0.2f * xv.w * xv.w);

    float q0 = A0 * iD0, q1 = A1 * iD1, q2 = A2 * iD2, q3 = A3 * iD3;
    float s0 = f0 * iD0, s1 = f1 * iD1, s2 = f2 * iD2, s3 = f3 * iD3;

    float Ar = A0 * s0 + A1 * s1 + A2 * s2 + A3 * s3;
    float Aq = A0 * q0 + A1 * q1 + A2 * q2 + A3 * q3;
    float coef = 0.1f * Ar * fast_rcp(1.f + 0.1f * Aq);

    v4f acc;
    acc.x = s0 - coef * q0;
    acc.y = s1 - coef * q1;
    acc.z = s2 - coef * q2;
    acc.w = s3 - coef * q3;

    // Per-lane b128 store: 512B contiguous per wave.
    *(v4f*)(out + (size_t)pid * 4) = acc;
  }
}

extern "C" void kernel_launch(void* const* d_in, const int* in_sizes, int n_in,
                              void* d_out, int out_size, void* d_ws,
                              size_t ws_size, hipStream_t stream) {
  (void)n_in; (void)out_size; (void)d_ws; (void)ws_size;
  const float* y = (const float*)d_in[0];
  const float* W = (const float*)d_in[1];
  const float* V = (const float*)d_in[2];
  float* out = (float*)d_out;

  const int B = in_sizes[0] / 8;      // 262144
  const int nTiles = B / 32;          // 32 particles per wave-iteration
  const int wavesPerBlock = 256 / 32; // 8 waves per block
  int blocks = (nTiles + wavesPerBlock - 1) / wavesPerBlock;
  if (blocks > 4096) blocks = 4096;   // grid-stride loop covers the rest
  if (blocks < 1) blocks = 1;

  relativistic_lagrangian_kernel<<<blocks, 256, 0, stream>>>(y, W, V, out,
                                                             nTiles);
}